// GATConvNodeLayer_77257871720739
// MI455X (gfx1250) — compile-verified
//
#include <hip/hip_runtime.h>
#include <math.h>

// GAT layer for MI455X (gfx1250, wave32).
//   x = feats @ W            -> V_WMMA_F32_16X16X4_F32 (fp32-exact, BW-bound anyway)
//   edge softmax             -> ordered-uint atomicMax + atomicAdd scatter
//   aggregation              -> 16 lanes/edge float4 gather + fp32 atomics (L2-resident acc)
//   BN(+ReLU)                -> two-pass column stats (bias cancels under mean-subtraction)

#define NN 100000
#define NE 3200000
#define D  64
#define NEG_SLOPE 0.2f
#define BN_EPS 1e-5f

typedef float v2f __attribute__((ext_vector_type(2)));
typedef float v8f __attribute__((ext_vector_type(8)));

// ---------- monotone float<->uint mapping for atomicMax on floats ----------
__device__ __forceinline__ unsigned f2ord(float f) {
    unsigned u = __float_as_uint(f);
    return (u >> 31) ? ~u : (u | 0x80000000u);
}
__device__ __forceinline__ float ord2f(unsigned u) {
    return (u & 0x80000000u) ? __uint_as_float(u & 0x7fffffffu) : __uint_as_float(~u);
}

// ---------- init workspace ----------
__global__ void gat_init(float* __restrict__ acc, unsigned* __restrict__ m_bits,
                         float* __restrict__ ssum, float* __restrict__ colsum,
                         float* __restrict__ colsq) {
    long i = (long)blockIdx.x * blockDim.x + threadIdx.x;
    if (i < (long)NN * D) acc[i] = 0.0f;
    if (i < NN) { m_bits[i] = 0u; ssum[i] = 0.0f; }
    if (i < D)      colsum[i] = 0.0f;
    else if (i < 2 * D) colsq[i - D] = 0.0f;
}

// ---------- X = feats @ W via V_WMMA_F32_16X16X4_F32 ----------
// 8 waves/block; wave w: row tile = blk*2 + (w>>2), col tile = w&3.
// A 16x4 layout : lane&15 = M, VGPR0/1 = K = (lane>>4)*2 + {0,1}
// B 4x16 layout : lane&15 = N, VGPR0/1 = K = (lane>>4)*2 + {0,1}
// C 16x16 layout: lane<16 -> rows r (VGPR r), lane>=16 -> rows r+8; col = lane&15
__global__ __launch_bounds__(256) void gat_gemm_wmma(const float* __restrict__ A,
                                                     const float* __restrict__ W,
                                                     float* __restrict__ X) {
    const int lane = threadIdx.x & 31;
    const int wave = threadIdx.x >> 5;
    const int row0 = (blockIdx.x * 2 + (wave >> 2)) * 16;
    const int col0 = (wave & 3) * 16;
    const int l15  = lane & 15;
    const int kb   = (lane >> 4) * 2;

    const float* __restrict__ arow = A + (size_t)(row0 + l15) * D;
    v8f c = {};
#pragma unroll
    for (int k0 = 0; k0 < D; k0 += 4) {
        v2f a = *(const v2f*)(arow + k0 + kb);          // 8B-aligned pair
        v2f b;
        b.x = W[(k0 + kb)     * D + col0 + l15];
        b.y = W[(k0 + kb + 1) * D + col0 + l15];
        c = __builtin_amdgcn_wmma_f32_16x16x4_f32(false, a, false, b,
                                                  (short)0, c, false, false);
    }
    const int colw  = col0 + l15;
    const int rbase = row0 + ((lane >> 4) << 3);
#pragma unroll
    for (int r = 0; r < 8; ++r)
        X[(size_t)(rbase + r) * D + colw] = c[r];
}

// ---------- per-node attention logits ----------
__global__ void gat_logits(const float* __restrict__ X, const float* __restrict__ att_src,
                           const float* __restrict__ att_dst, float* __restrict__ a_src,
                           float* __restrict__ a_dst) {
    __shared__ float ss[D], sd[D];
    if (threadIdx.x < D) { ss[threadIdx.x] = att_src[threadIdx.x];
                           sd[threadIdx.x] = att_dst[threadIdx.x]; }
    __syncthreads();
    int n = blockIdx.x * blockDim.x + threadIdx.x;
    if (n >= NN) return;
    const float* xr = X + (size_t)n * D;
    float s = 0.0f, d = 0.0f;
#pragma unroll 8
    for (int k = 0; k < D; ++k) { float v = xr[k]; s += v * ss[k]; d += v * sd[k]; }
    a_src[n] = s; a_dst[n] = d;
}

// ---------- edge pass 1: leaky-relu logit, scatter max ----------
__global__ void gat_edge_max(const int* __restrict__ ei, const float* __restrict__ a_src,
                             const float* __restrict__ a_dst, float* __restrict__ e_buf,
                             unsigned* __restrict__ m_bits) {
    long id = (long)blockIdx.x * blockDim.x + threadIdx.x;
    if (id >= (long)NE + NN) return;
    int s, d;
    if (id < NE) { s = ei[id]; d = ei[NE + id]; }
    else         { s = d = (int)(id - NE); }          // self loops
    float v = a_src[s] + a_dst[d];
    float e = v > 0.0f ? v : NEG_SLOPE * v;
    e_buf[id] = e;
    atomicMax(&m_bits[d], f2ord(e));
}

// ---------- edge pass 2: exp(e - max), scatter sum ----------
__global__ void gat_edge_expsum(const int* __restrict__ ei, const unsigned* __restrict__ m_bits,
                                float* __restrict__ e_buf, float* __restrict__ ssum) {
    long id = (long)blockIdx.x * blockDim.x + threadIdx.x;
    if (id >= (long)NE + NN) return;
    int d = (id < NE) ? ei[NE + id] : (int)(id - NE);
    float ex = expf(e_buf[id] - ord2f(m_bits[d]));
    e_buf[id] = ex;
    atomicAdd(&ssum[d], ex);
}

// ---------- edge pass 3: alpha-weighted scatter of x[src]; 16 lanes per edge ----------
__global__ void gat_edge_agg(const int* __restrict__ ei, const float* __restrict__ e_buf,
                             const float* __restrict__ ssum, const float* __restrict__ X,
                             float* __restrict__ acc) {
    long t = (long)blockIdx.x * blockDim.x + threadIdx.x;
    if (t >= ((long)NE + NN) * 16) return;
    long id = t >> 4;
    int  q  = (int)(t & 15);
    int s, d;
    if (id < NE) { s = ei[id]; d = ei[NE + id]; }
    else         { s = d = (int)(id - NE); }
    float alpha = e_buf[id] / ssum[d];
    float4 v = ((const float4*)(X + (size_t)s * D))[q];
    float* o = acc + (size_t)d * D + q * 4;
    atomicAdd(o + 0, v.x * alpha);
    atomicAdd(o + 1, v.y * alpha);
    atomicAdd(o + 2, v.z * alpha);
    atomicAdd(o + 3, v.w * alpha);
}

// ---------- per-channel sum / sum-of-squares ----------
__global__ __launch_bounds__(256) void gat_colstats(const float* __restrict__ acc,
                                                    float* __restrict__ colsum,
                                                    float* __restrict__ colsq) {
    __shared__ float s1[256], s2[256];
    const int d = threadIdx.x & 63;
    const int r = threadIdx.x >> 6;
    float p1 = 0.0f, p2 = 0.0f;
    for (int n = blockIdx.x * 4 + r; n < NN; n += gridDim.x * 4) {
        float v = acc[(size_t)n * D + d];
        p1 += v; p2 += v * v;
    }
    s1[threadIdx.x] = p1; s2[threadIdx.x] = p2;
    __syncthreads();
    if (r == 0) {
        p1 = s1[d] + s1[d + 64] + s1[d + 128] + s1[d + 192];
        p2 = s2[d] + s2[d + 64] + s2[d + 128] + s2[d + 192];
        atomicAdd(&colsum[d], p1);
        atomicAdd(&colsq[d],  p2);
    }
}

// ---------- batchnorm (training stats) + ReLU; bias cancels under mean subtraction ----------
__global__ void gat_bn_relu(const float* __restrict__ acc, const float* __restrict__ colsum,
                            const float* __restrict__ colsq, const float* __restrict__ gamma,
                            const float* __restrict__ beta, float* __restrict__ out) {
    long i = (long)blockIdx.x * blockDim.x + threadIdx.x;
    if (i >= (long)NN * D) return;
    int dch = (int)(i & 63);
    float mu  = colsum[dch] * (1.0f / NN);
    float var = colsq[dch] * (1.0f / NN) - mu * mu;
    float v = (acc[i] - mu) * rsqrtf(var + BN_EPS) * gamma[dch] + beta[dch];
    out[i] = v > 0.0f ? v : 0.0f;
}

extern "C" void kernel_launch(void* const* d_in, const int* in_sizes, int n_in,
                              void* d_out, int out_size, void* d_ws, size_t ws_size,
                              hipStream_t stream) {
    const float* feats = (const float*)d_in[0];
    const int*   ei    = (const int*)  d_in[1];   // (2,E) int32 — src row then dst row
    const float* Wm    = (const float*)d_in[2];
    const float* attS  = (const float*)d_in[3];
    const float* attD  = (const float*)d_in[4];
    const float* gamma = (const float*)d_in[6];
    const float* beta  = (const float*)d_in[7];
    float* out = (float*)d_out;
    (void)in_sizes; (void)n_in; (void)out_size; (void)ws_size;

    // workspace carve-up (~66 MB)
    float*    ws     = (float*)d_ws;
    float*    X      = ws;                               // N*D
    float*    acc    = X + (size_t)NN * D;               // N*D
    float*    e_buf  = acc + (size_t)NN * D;             // E+N
    float*    a_src  = e_buf + (NE + NN);                // N
    float*    a_dst  = a_src + NN;                       // N
    unsigned* m_bits = (unsigned*)(a_dst + NN);          // N
    float*    ssum   = (float*)(m_bits + NN);            // N
    float*    colsum = ssum + NN;                        // 64
    float*    colsq  = colsum + D;                       // 64

    const long ND   = (long)NN * D;
    const long EDGE = (long)NE + NN;

    gat_init<<<(int)((ND + 255) / 256), 256, 0, stream>>>(acc, m_bits, ssum, colsum, colsq);
    gat_gemm_wmma<<<NN / 32, 256, 0, stream>>>(feats, Wm, X);
    gat_logits<<<(NN + 255) / 256, 256, 0, stream>>>(X, attS, attD, a_src, a_dst);
    gat_edge_max<<<(int)((EDGE + 255) / 256), 256, 0, stream>>>(ei, a_src, a_dst, e_buf, m_bits);
    gat_edge_expsum<<<(int)((EDGE + 255) / 256), 256, 0, stream>>>(ei, m_bits, e_buf, ssum);
    gat_edge_agg<<<(int)((EDGE * 16 + 255) / 256), 256, 0, stream>>>(ei, e_buf, ssum, X, acc);
    gat_colstats<<<256, 256, 0, stream>>>(acc, colsum, colsq);
    gat_bn_relu<<<(int)((ND + 255) / 256), 256, 0, stream>>>(acc, colsum, colsq, gamma, beta, out);
}